// MonNet_52888227283699
// MI455X (gfx1250) — compile-verified
//
#include <hip/hip_runtime.h>

typedef __attribute__((ext_vector_type(2))) float v2f;
typedef __attribute__((ext_vector_type(4))) float v4f;
typedef __attribute__((ext_vector_type(8))) float v8f;

#define DHID   64
#define BATCH  131072
#define NIN    128
#define NOUT   8
#define SW1    132   // LDS row stride for W1 (132 % 64 == 4 -> conflict-free b64 A loads)
#define SM     68    // LDS row stride for 64x64 matrices (68 % 64 == 4)
#define TOLF   1e-5f
#define MAXIT  50

// ---------------------------------------------------------------------------
// f32 WMMA: D(16x16) = A(16x4) * B(4x16) + C   -> v_wmma_f32_16x16x4_f32
// ---------------------------------------------------------------------------
__device__ __forceinline__ v8f wmma4(v2f a, v2f b, v8f c) {
  return __builtin_amdgcn_wmma_f32_16x16x4_f32(false, a, false, b, (short)0, c,
                                               false, false);
}

__device__ __forceinline__ v8f vzero8() {
  v8f z;
#pragma unroll
  for (int v = 0; v < 8; ++v) z[v] = 0.0f;
  return z;
}

// ---------------------------------------------------------------------------
// outT[16x64] = Alds[64x64] @ ActT[64x16] + cin
//   Act (16 rows x 64 cols) lives in registers in WMMA C/D ("transposed-D")
//   layout: tile t VGPR v: lanes 0-15 -> Act[lane][16t+v],
//                          lanes 16-31 -> Act[lane-16][16t+v+8].
//   Conversion to the 4x16 B-operand layout is a XOR-16 lane swap + select.
//   Tile-outer structure keeps only 8 shuffle temps live at a time.
//   A operand streamed from LDS: one ds_load_b64 per WMMA, conflict-free
//   (row stride == 4 mod 64 dwords).
// ---------------------------------------------------------------------------
__device__ __forceinline__ void mm64(const float* Alds, const v8f* src,
                                     const v8f* cin, v8f* dst,
                                     const int r, const int koff,
                                     const bool hi) {
  v8f acc[4];
#pragma unroll
  for (int m = 0; m < 4; ++m) acc[m] = cin[m];

#pragma unroll
  for (int t = 0; t < 4; ++t) {
    float s[8];
#pragma unroll
    for (int v = 0; v < 8; ++v) s[v] = __shfl_xor(src[t][v], 16, 32);
#pragma unroll
    for (int j = 0; j < 4; ++j) {
      const int kt = t * 4 + j;
      v2f b;
      if (j == 0) {
        b.x = hi ? s[2] : src[t][0];
        b.y = hi ? s[3] : src[t][1];
      } else if (j == 1) {
        b.x = hi ? s[6] : src[t][4];
        b.y = hi ? s[7] : src[t][5];
      } else if (j == 2) {
        b.x = hi ? src[t][2] : s[0];
        b.y = hi ? src[t][3] : s[1];
      } else {
        b.x = hi ? src[t][6] : s[4];
        b.y = hi ? src[t][7] : s[5];
      }
#pragma unroll
      for (int m = 0; m < 4; ++m) {
        v2f a = *(const v2f*)(Alds + (m * 16 + r) * SM + 4 * kt + koff);
        acc[m] = wmma4(a, b, acc[m]);
      }
    }
  }
#pragma unroll
  for (int m = 0; m < 4; ++m) dst[m] = acc[m];
}

// ---------------------------------------------------------------------------
// Setup: W = (1-m)I - Aw^T Aw + Bw - Bw^T ; Winv = inv(2I - W)
// Single block, Gauss-Jordan with partial pivoting on [64 | I] in LDS.
// ---------------------------------------------------------------------------
__global__ __launch_bounds__(128) void monnet_setup(
    const float* __restrict__ Aw, const float* __restrict__ Bw,
    float* __restrict__ Wout, float* __restrict__ Winv) {
  __shared__ float aug[64][130];
  __shared__ float colk[64];
  __shared__ int piv;
  const int t = threadIdx.x;  // 0..127, owns augmented column t

  for (int e = t; e < 4096; e += 128) {
    const int i = e >> 6, j = e & 63;
    float s = 0.0f;
    for (int k = 0; k < 64; ++k) s += Aw[k * 64 + i] * Aw[k * 64 + j];
    const float w = ((i == j) ? 0.9f : 0.0f) - s + Bw[i * 64 + j] - Bw[j * 64 + i];
    Wout[e] = w;
    aug[i][j] = ((i == j) ? 2.0f : 0.0f) - w;  // (1+alpha)I - alpha*W, alpha=1
    aug[i][64 + j] = (i == j) ? 1.0f : 0.0f;
  }
  __syncthreads();

  for (int k = 0; k < 64; ++k) {
    if (t == 0) {
      int p = k;
      float mv = fabsf(aug[k][k]);
      for (int i2 = k + 1; i2 < 64; ++i2) {
        const float a = fabsf(aug[i2][k]);
        if (a > mv) { mv = a; p = i2; }
      }
      piv = p;
    }
    __syncthreads();
    const int p = piv;
    if (p != k) {
      const float tmp = aug[k][t];
      aug[k][t] = aug[p][t];
      aug[p][t] = tmp;
    }
    __syncthreads();
    if (t < 64) colk[t] = aug[t][k];
    __syncthreads();
    const float pinv = 1.0f / colk[k];
    aug[k][t] *= pinv;               // scale pivot row (own column only)
    const float rk = aug[k][t];
    for (int i2 = 0; i2 < 64; ++i2)  // eliminate (own column only)
      if (i2 != k) aug[i2][t] -= colk[i2] * rk;
    __syncthreads();
  }
  for (int e = t; e < 4096; e += 128) Winv[e] = aug[e >> 6][64 + (e & 63)];
}

// ---------------------------------------------------------------------------
// Main: 8 waves/block, each wave owns 16 rows kept fully register-resident
// through the Peaceman-Rachford loop. 1024 blocks x 128 rows = 131072.
// ---------------------------------------------------------------------------
__global__ __launch_bounds__(256) void monnet_main(
    const float* __restrict__ x, const float* __restrict__ W1,
    const float* __restrict__ b1, const float* __restrict__ Uw,
    const float* __restrict__ Ub, const float* __restrict__ Pw,
    const float* __restrict__ Pb, const float* __restrict__ Wm,
    const float* __restrict__ Wi, float* __restrict__ out) {
  __shared__ float sW1[DHID * SW1];
  __shared__ float sU[DHID * SM];
  __shared__ float sWi[DHID * SM];
  __shared__ float sWm[DHID * SM];
  __shared__ float sPwT[DHID * NOUT];
  __shared__ float sBias[8 * 32 * 32];  // per-wave parking for bias tiles
  __shared__ float sb1[DHID];
  __shared__ float sUb[DHID];
  __shared__ float sPb[NOUT];

  const int tid = threadIdx.x;
  for (int i = tid; i < DHID * NIN; i += 256)
    sW1[(i >> 7) * SW1 + (i & 127)] = W1[i];
  for (int i = tid; i < DHID * DHID; i += 256) {
    const int r0 = i >> 6, c0i = i & 63;
    sU[r0 * SM + c0i] = Uw[i];
    sWi[r0 * SM + c0i] = Wi[i];
    sWm[r0 * SM + c0i] = Wm[i];
  }
  for (int i = tid; i < NOUT * DHID; i += 256)
    sPwT[(i & 63) * NOUT + (i >> 6)] = Pw[i];
  if (tid < DHID) { sb1[tid] = b1[tid]; sUb[tid] = Ub[tid]; }
  if (tid < NOUT) sPb[tid] = Pb[tid];
  __syncthreads();

  const int lane = tid & 31;
  const int wave = tid >> 5;
  const int r = lane & 15;
  const bool hi = lane >= 16;
  const int koff = hi ? 2 : 0;
  const int bshift = hi ? 8 : 0;
  const int row = (blockIdx.x * 8 + wave) * 16 + r;
  float* myBias = sBias + (wave * 32) * 32 + lane;  // [wave][elem][lane]

  // ---- pre layer: hT = W1 @ xT + b1 ; relu -----------------------------
  v8f h[4];
#pragma unroll
  for (int m = 0; m < 4; ++m)
#pragma unroll
    for (int v = 0; v < 8; ++v) h[m][v] = sb1[m * 16 + v + bshift];

  const float* xrow = x + (size_t)row * NIN;
#pragma unroll 8
  for (int kt = 0; kt < 32; ++kt) {
    v2f b = *(const v2f*)(xrow + 4 * kt + koff);
#pragma unroll
    for (int m = 0; m < 4; ++m) {
      v2f a = *(const v2f*)(sW1 + (m * 16 + r) * SW1 + 4 * kt + koff);
      h[m] = wmma4(a, b, h[m]);
    }
  }
#pragma unroll
  for (int m = 0; m < 4; ++m)
#pragma unroll
    for (int v = 0; v < 8; ++v) h[m][v] = fmaxf(h[m][v], 0.0f);

  // ---- bias = h @ Uw^T + Ub ; c0 = bias @ Winv^T (loop-invariant) ------
  v8f ci[4], bias[4], c0[4], z12[4];
#pragma unroll
  for (int m = 0; m < 4; ++m)
#pragma unroll
    for (int v = 0; v < 8; ++v) ci[m][v] = sUb[m * 16 + v + bshift];
  mm64(sU, h, ci, bias, r, koff, hi);

#pragma unroll
  for (int m = 0; m < 4; ++m) ci[m] = vzero8();
  mm64(sWi, bias, ci, c0, r, koff, hi);

  // park bias in LDS so it is not live through the PR loop (coalesced b32)
#pragma unroll
  for (int m = 0; m < 4; ++m)
#pragma unroll
    for (int v = 0; v < 8; ++v) myBias[(m * 8 + v) * 32] = bias[m][v];

  // ---- Peaceman-Rachford loop, register-resident, per-wave early exit --
  v8f z[4], u[4];
#pragma unroll
  for (int m = 0; m < 4; ++m) { z[m] = vzero8(); u[m] = vzero8(); }

#pragma unroll 1
  for (int it = 0; it < MAXIT; ++it) {
    // u12 = 2z - u (stored in u)
#pragma unroll
    for (int m = 0; m < 4; ++m)
#pragma unroll
      for (int v = 0; v < 8; ++v) u[m][v] = 2.0f * z[m][v] - u[m][v];

    // z12 = (bias + u12) @ Winv^T = c0 + u12 @ Winv^T
    mm64(sWi, u, c0, z12, r, koff, hi);

    if ((it & 1) == 0) {
      // cheap iteration: no convergence bookkeeping
#pragma unroll
      for (int m = 0; m < 4; ++m)
#pragma unroll
        for (int v = 0; v < 8; ++v) {
          const float un = 2.0f * z12[m][v] - u[m][v];
          u[m][v] = un;
          z[m][v] = fmaxf(un, 0.0f);
        }
    } else {
      float s1 = 0.0f, s2 = 0.0f;
#pragma unroll
      for (int m = 0; m < 4; ++m)
#pragma unroll
        for (int v = 0; v < 8; ++v) {
          const float un = 2.0f * z12[m][v] - u[m][v];
          u[m][v] = un;
          const float zn = fmaxf(un, 0.0f);
          const float d = zn - z[m][v];
          s1 += d * d;
          s2 += zn * zn;
          z[m][v] = zn;
        }
#pragma unroll
      for (int off = 16; off > 0; off >>= 1) {
        s1 += __shfl_xor(s1, off, 32);
        s2 += __shfl_xor(s2, off, 32);
      }
      const int done = (sqrtf(s1) <= TOLF * (1e-6f + sqrtf(s2))) ? 1 : 0;
      if (__builtin_amdgcn_readfirstlane(done)) break;
    }
  }

  // ---- zn = relu(z @ W^T + bias) ---------------------------------------
#pragma unroll
  for (int m = 0; m < 4; ++m)
#pragma unroll
    for (int v = 0; v < 8; ++v) ci[m][v] = myBias[(m * 8 + v) * 32];

  v8f zn4[4];
  mm64(sWm, z, ci, zn4, r, koff, hi);
#pragma unroll
  for (int m = 0; m < 4; ++m)
#pragma unroll
    for (int v = 0; v < 8; ++v) zn4[m][v] = fmaxf(zn4[m][v], 0.0f);

  // ---- post: out = zn @ Pw^T + Pb  (8-wide, VALU + XOR-16 reduce) ------
  float po[NOUT];
#pragma unroll
  for (int o = 0; o < NOUT; ++o) po[o] = 0.0f;
#pragma unroll
  for (int m = 0; m < 4; ++m)
#pragma unroll
    for (int v = 0; v < 8; ++v) {
      const int c = m * 16 + v + bshift;
      const float zv = zn4[m][v];
      const float* pp = sPwT + c * NOUT;
#pragma unroll
      for (int o = 0; o < NOUT; ++o) po[o] += zv * pp[o];
    }
#pragma unroll
  for (int o = 0; o < NOUT; ++o) po[o] += __shfl_xor(po[o], 16, 32);

  if (!hi) {
    float* op = out + (size_t)row * NOUT;
    v4f lo4, hi4;
#pragma unroll
    for (int o = 0; o < 4; ++o) lo4[o] = po[o] + sPb[o];
#pragma unroll
    for (int o = 0; o < 4; ++o) hi4[o] = po[o + 4] + sPb[o + 4];
    *(v4f*)(op) = lo4;
    *(v4f*)(op + 4) = hi4;
  }
}

// ---------------------------------------------------------------------------
extern "C" void kernel_launch(void* const* d_in, const int* in_sizes, int n_in,
                              void* d_out, int out_size, void* d_ws,
                              size_t ws_size, hipStream_t stream) {
  const float* x  = (const float*)d_in[0];
  const float* W1 = (const float*)d_in[1];
  const float* b1 = (const float*)d_in[2];
  const float* Uw = (const float*)d_in[3];
  const float* Ub = (const float*)d_in[4];
  const float* Aw = (const float*)d_in[5];
  const float* Bw = (const float*)d_in[6];
  const float* Pw = (const float*)d_in[7];
  const float* Pb = (const float*)d_in[8];

  float* Wm = (float*)d_ws;        // 64x64 W
  float* Wi = Wm + DHID * DHID;    // 64x64 Winv

  monnet_setup<<<1, 128, 0, stream>>>(Aw, Bw, Wm, Wi);
  monnet_main<<<BATCH / 128, 256, 0, stream>>>(x, W1, b1, Uw, Ub, Pw, Pb, Wm,
                                               Wi, (float*)d_out);
}